// HBVMulTDET_33758442946688
// MI455X (gfx1250) — compile-verified
//
#include <hip/hip_runtime.h>
#include <cstdint>
#include <cstddef>

// HBV model constants (fixed in the reference module)
#define NSTEP 730
#define NGRID 1000
#define MU    8
#define GPB   4             // grids per block
#define BLOCK (GPB * MU)    // 32 threads = 1 wave32 -> barrier-free pipeline
#define PRECS 1e-5f

__constant__ float c_LB[13] = {1.0f, 50.0f,  0.05f, 0.01f, 0.001f, 0.2f, 0.0f,
                               0.0f, -2.5f,  0.5f,  0.0f,  0.0f,   0.3f};
__constant__ float c_UB[13] = {6.0f, 1000.0f, 0.9f, 0.5f,  0.2f,   1.0f, 10.0f,
                               100.0f, 2.5f, 10.0f, 0.1f,  0.2f,   5.0f};

// ---- CDNA5 async global->LDS staging ------------------------------------
typedef __attribute__((address_space(1))) int as1_int;
typedef __attribute__((address_space(3))) int as3_int;

#if __has_builtin(__builtin_amdgcn_global_load_async_to_lds_b32)
__device__ __forceinline__ void stage_b32(const float* g, float* l) {
  __builtin_amdgcn_global_load_async_to_lds_b32(
      (as1_int*)(uintptr_t)g,
      (as3_int*)(uint32_t)(uintptr_t)l,
      /*offset=*/0, /*cpol=*/0);
}
#else
#warning "gfx1250 async-to-LDS builtin unavailable: using synchronous LDS staging fallback"
__device__ __forceinline__ void stage_b32(const float* g, float* l) {
  *l = __ldg(g);
}
#endif

__device__ __forceinline__ void wait_async0() {
#if __has_builtin(__builtin_amdgcn_s_wait_asynccnt)
  __builtin_amdgcn_s_wait_asynccnt(0);
#else
  asm volatile("s_wait_asynccnt 0" ::: "memory");
#endif
  // zero-instruction compiler fence: keep LDS reads below the wait
  asm volatile("" ::: "memory");
}
// -------------------------------------------------------------------------

__global__ __launch_bounds__(BLOCK) void hbv_scan_kernel(
    const float* __restrict__ x,        // (NSTEP, NGRID, 3)
    const float* __restrict__ par,      // (NSTEP, NGRID, 13, MU)
    const int*   __restrict__ staind_p, // scalar
    const int*   __restrict__ tdlst,    // (nlist,)
    int nlist,
    float* __restrict__ out)            // (NSTEP, NGRID, 1)
{
  __shared__ float xbuf[2][GPB * 3];        // P,T,ET per grid
  __shared__ float pbuf[2][GPB * 13 * MU];  // per-step params (only masked slots valid)

  const int tid = threadIdx.x;
  const int gl  = tid >> 3;        // grid-local 0..GPB-1
  const int m   = tid & 7;         // mu lane 0..7
  const int g0  = blockIdx.x * GPB;
  const int g   = g0 + gl;
  const int gc  = (g < NGRID) ? g : (NGRID - 1);   // clamp for safe loads

  // time-varying parameter mask from tdlst: idx = (tdlst-1) mod 13 (python mod)
  int mask = 0;
  for (int i = 0; i < nlist; ++i) {
    int r = (tdlst[i] - 1) % 13;
    if (r < 0) r += 13;
    mask |= (1 << r);
  }

  const int staind = staind_p[0];

  // Pre-scaled time-invariant parameters (from step `staind`)
  float baseT[13];
  {
    const size_t bofs = ((size_t)staind * NGRID + gc) * (13 * MU) + m;
#pragma unroll
    for (int p = 0; p < 13; ++p)
      baseT[p] = c_LB[p] + par[bofs + (size_t)p * MU] * (c_UB[p] - c_LB[p]);
  }

  // Issue async loads of step-t inputs into LDS buffer b
  auto stage = [&](int t, int b) {
    if (tid < GPB * 3) {   // lanes 0..11 stage P,T,ET (EXEC-predicated)
      const int sgl = tid / 3, c = tid - sgl * 3;
      int sg = g0 + sgl; if (sg >= NGRID) sg = NGRID - 1;
      stage_b32(x + ((size_t)t * NGRID + sg) * 3 + c, &xbuf[b][tid]);
    }
    const size_t prow = ((size_t)t * NGRID + gc) * (13 * MU) + m;
#pragma unroll
    for (int p = 0; p < 13; ++p)
      if (mask & (1 << p))   // uniform branch: one async b32 per thread per varying param
        stage_b32(par + prow + (size_t)p * MU, &pbuf[b][(gl * 13 + p) * MU + m]);
  };

  stage(0, 0);  // prologue: pipeline fill

  float SP = 0.001f, MW = 0.001f, SM = 0.001f, SUZ = 0.001f, SLZ = 0.001f;
  const bool writer = (m == 0) && (g < NGRID);

  for (int t = 0; t < NSTEP; ++t) {
    wait_async0();              // this wave's async LDS writes for buffer t&1 landed
    if (BLOCK > warpSize) __syncthreads();   // compile-time removed for 1-wave blocks
    if (t + 1 < NSTEP) stage(t + 1, (t + 1) & 1);

    // warm L2 ahead of the async pipeline (global_prefetch_b8, counter-free)
    if (t + 8 < NSTEP) {
      __builtin_prefetch(x + ((size_t)(t + 8) * NGRID + gc) * 3, 0, 1);
      __builtin_prefetch(par + ((size_t)(t + 8) * NGRID + gc) * (13 * MU) + m, 0, 1);
    }

    const int cb = t & 1;
    const float P  = xbuf[cb][gl * 3 + 0];
    const float T  = xbuf[cb][gl * 3 + 1];
    const float ET = xbuf[cb][gl * 3 + 2];

    float pv[13];
#pragma unroll
    for (int p = 0; p < 13; ++p)
      pv[p] = (mask & (1 << p))
                ? c_LB[p] + pbuf[cb][(gl * 13 + p) * MU + m] * (c_UB[p] - c_LB[p])
                : baseT[p];

    const float BETA = pv[0],  FC    = pv[1],  K0  = pv[2],  K1   = pv[3];
    const float K2   = pv[4],  LP    = pv[5],  PERCc = pv[6], UZL = pv[7];
    const float TT   = pv[8],  CFMAX = pv[9],  CFR = pv[10], CWH  = pv[11];
    const float BETAET = pv[12];

    const float is_rain = (T >= TT) ? 1.0f : 0.0f;
    const float RAIN = P * is_rain;
    const float SNOW = P - RAIN;
    SP += SNOW;
    const float melt = fminf(fmaxf(CFMAX * (T - TT), 0.0f), SP);
    MW += melt;  SP -= melt;
    const float refr = fminf(fmaxf(CFR * CFMAX * (TT - T), 0.0f), MW);
    SP += refr;  MW -= refr;
    const float tosoil = fmaxf(MW - CWH * SP, 0.0f);
    MW -= tosoil;
    const float soil_wetness = fminf(fmaxf(powf(SM / FC, BETA), 0.0f), 1.0f);
    const float recharge = (RAIN + tosoil) * soil_wetness;
    SM += RAIN + tosoil - recharge;
    const float excess = fmaxf(SM - FC, 0.0f);
    SM -= excess;
    const float evapfactor = fminf(fmaxf(powf(SM / (LP * FC), BETAET), 0.0f), 1.0f);
    const float ETact = fminf(SM, ET * evapfactor);
    SM = fmaxf(SM - ETact, PRECS);
    SUZ += recharge + excess;
    const float PERC = fminf(SUZ, PERCc);
    SUZ -= PERC;
    const float Q0 = K0 * fmaxf(SUZ - UZL, 0.0f);
    SUZ -= Q0;
    const float Q1 = K1 * SUZ;
    SUZ -= Q1;
    SLZ += PERC;
    const float Q2 = K2 * SLZ;
    SLZ -= Q2;

    // mean over mu (8 adjacent lanes of a wave32)
    float s = Q0 + Q1 + Q2;
    s += __shfl_xor(s, 1, 8);
    s += __shfl_xor(s, 2, 8);
    s += __shfl_xor(s, 4, 8);
    if (writer) out[(size_t)t * NGRID + g] = s * 0.125f;
  }
}

extern "C" void kernel_launch(void* const* d_in, const int* in_sizes, int n_in,
                              void* d_out, int out_size, void* d_ws, size_t ws_size,
                              hipStream_t stream) {
  (void)n_in; (void)out_size; (void)d_ws; (void)ws_size;
  const float* x      = (const float*)d_in[0];
  const float* par    = (const float*)d_in[1];
  const int*   staind = (const int*)d_in[2];
  const int*   tdlst  = (const int*)d_in[3];
  const int    nlist  = in_sizes[3];

  dim3 grid((NGRID + GPB - 1) / GPB);   // 250 single-wave workgroups
  dim3 block(BLOCK);
  hipLaunchKernelGGL(hbv_scan_kernel, grid, block, 0, stream,
                     x, par, staind, tdlst, nlist, (float*)d_out);
}